// RGCN_82025285419624
// MI455X (gfx1250) — compile-verified
//
#include <hip/hip_runtime.h>
#include <cstddef>

#define N_NODES 50000
#define N_REL   8
#define HID     128
#define NCLS    40
#define N_EDGE  800000

typedef __attribute__((ext_vector_type(2))) float v2f;
typedef __attribute__((ext_vector_type(8))) float v8f;

__device__ __forceinline__ v8f wmma_f32_16x16x4(v2f a, v2f b, v8f c) {
#if defined(__gfx1250__)
  // D = A(16x4 f32) * B(4x16 f32) + C(16x16 f32)
  return __builtin_amdgcn_wmma_f32_16x16x4_f32(false, a, false, b, (short)0, c,
                                               false, false);
#else
  (void)a; (void)b;
  return c;
#endif
}

// ---------------- counting / sorting ----------------

__global__ void k_count(const int* __restrict__ ei, const int* __restrict__ et,
                        int* __restrict__ cnt, int* __restrict__ relCnt) {
  int e = blockIdx.x * blockDim.x + threadIdx.x;
  if (e >= N_EDGE) return;
  int dst = ei[N_EDGE + e];
  int r = et[e];
  atomicAdd(&cnt[dst * N_REL + r], 1);
  atomicAdd(&relCnt[r], 1);
}

__global__ void k_prefix(const int* __restrict__ relCnt, int* __restrict__ relOff) {
  if (threadIdx.x == 0 && blockIdx.x == 0) {
    int off = 0;
    for (int r = 0; r < N_REL; ++r) {
      relOff[r] = off;
      off += (relCnt[r] + 15) & ~15;   // pad each relation segment to 16 edges
    }
    relOff[N_REL] = off;
  }
}

__global__ void k_perm(const int* __restrict__ et, const int* __restrict__ relOff,
                       int* __restrict__ relFill, int* __restrict__ perm) {
  int e = blockIdx.x * blockDim.x + threadIdx.x;
  if (e >= N_EDGE) return;
  int r = et[e];
  int pos = relOff[r] + atomicAdd(&relFill[r], 1);
  perm[pos] = e;   // padding slots stay -1 (preset via memset 0xFF)
}

// ---------------- layer 1: gather-scale-scatter ----------------
// one wave per edge; lane handles 4 contiguous floats (float4 gather from W1)
__global__ void k_l1_scatter(const int* __restrict__ ei, const int* __restrict__ et,
                             const float* __restrict__ W1, const int* __restrict__ cnt,
                             float* __restrict__ h) {
  long gid = (long)blockIdx.x * blockDim.x + threadIdx.x;
  int e = (int)(gid >> 5);
  int lane = (int)(gid & 31);
  if (e >= N_EDGE) return;
  int src = ei[e];
  int dst = ei[N_EDGE + e];
  int r = et[e];
  int c = cnt[dst * N_REL + r];
  float s = 1.0f / (float)(c > 0 ? c : 1);
  const float4* w = (const float4*)(W1 + ((size_t)r * N_NODES + src) * HID) + lane;
  float4 v = *w;
  float* o = h + (size_t)dst * HID + lane * 4;
  atomicAdd(o + 0, v.x * s);
  atomicAdd(o + 1, v.y * s);
  atomicAdd(o + 2, v.z * s);
  atomicAdd(o + 3, v.w * s);
}

__global__ void k_relu1(float* __restrict__ h, const float* __restrict__ root1,
                        const float* __restrict__ b1) {
  int i = blockIdx.x * blockDim.x + threadIdx.x;
  if (i >= N_NODES * HID) return;
  float v = h[i] + root1[i] + b1[i & (HID - 1)];
  h[i] = v > 0.f ? v : 0.f;
}

// ---------------- h2 = h @ root2 + b2  (initializes h2, dense WMMA GEMM) ----
// block = 256 threads = 8 waves; wave w owns output columns 16w..16w+15
__global__ __launch_bounds__(256) void k_root_gemm(const float* __restrict__ h,
                                                   const float* __restrict__ root2,
                                                   const float* __restrict__ b2,
                                                   float* __restrict__ h2) {
  __shared__ float As[16][HID];
  int row0 = blockIdx.x * 16;
  int tid = threadIdx.x;
  for (int idx = tid; idx < 16 * HID; idx += 256) {
    int m = idx >> 7, kk = idx & (HID - 1);
    As[m][kk] = h[(size_t)(row0 + m) * HID + kk];
  }
  __syncthreads();
  int wave = tid >> 5, lane = tid & 31;
  int half = lane >> 4, ln = lane & 15;
  int col = wave * 16 + ln;
  v8f acc = {};
  for (int k0 = 0; k0 < HID; k0 += 4) {
    int ka = k0 + half * 2;
    v2f a, b;
    a[0] = As[ln][ka];  a[1] = As[ln][ka + 1];
    b[0] = root2[(size_t)ka * HID + col];
    b[1] = root2[(size_t)(ka + 1) * HID + col];
    acc = wmma_f32_16x16x4(a, b, acc);
  }
  float bias = b2[col];
  for (int v = 0; v < 8; ++v) {
    int m = v + half * 8;
    h2[(size_t)(row0 + m) * HID + col] = acc[v] + bias;
  }
}

// ---------------- layer 2 messages: per-relation 16-edge GEMM tiles --------
// A = gathered h[src] rows (LDS), B = W2[r]; result rows scaled by 1/cnt and
// atomically scattered into h2. Padding rows (perm == -1) contribute zero.
__global__ __launch_bounds__(256) void k_edge_gemm(const float* __restrict__ h,
                                                   const float* __restrict__ W2,
                                                   const int* __restrict__ perm,
                                                   const int* __restrict__ relOff,
                                                   const int* __restrict__ cnt,
                                                   const int* __restrict__ ei,
                                                   const int* __restrict__ et,
                                                   float* __restrict__ h2) {
  __shared__ float As[16][HID];
  __shared__ int   sSrc[16];
  __shared__ int   sDst[16];
  __shared__ float sScale[16];
  int tileBase = blockIdx.x * 16;
  if (tileBase >= relOff[N_REL]) return;           // block-uniform exit
  int r = 0;
  while (r < N_REL - 1 && tileBase >= relOff[r + 1]) ++r;
  int tid = threadIdx.x;
  if (tid < 16) {
    int e = perm[tileBase + tid];
    int src = -1, dst = -1; float s = 0.f;
    if (e >= 0) {
      src = ei[e];
      dst = ei[N_EDGE + e];
      int c = cnt[dst * N_REL + r];
      s = 1.0f / (float)(c > 0 ? c : 1);
    }
    sSrc[tid] = src; sDst[tid] = dst; sScale[tid] = s;
  }
  __syncthreads();
  for (int idx = tid; idx < 16 * HID; idx += 256) {
    int m = idx >> 7, kk = idx & (HID - 1);
    int src = sSrc[m];
    As[m][kk] = (src >= 0) ? h[(size_t)src * HID + kk] : 0.f;
  }
  __syncthreads();
  int wave = tid >> 5, lane = tid & 31;
  int half = lane >> 4, ln = lane & 15;
  int col = wave * 16 + ln;
  const float* B = W2 + (size_t)r * HID * HID;     // B[k][n], n contiguous
  v8f acc = {};
  for (int k0 = 0; k0 < HID; k0 += 4) {
    int ka = k0 + half * 2;
    v2f a, b;
    a[0] = As[ln][ka];  a[1] = As[ln][ka + 1];
    b[0] = B[(size_t)ka * HID + col];
    b[1] = B[(size_t)(ka + 1) * HID + col];
    acc = wmma_f32_16x16x4(a, b, acc);
  }
  for (int v = 0; v < 8; ++v) {
    int m = v + half * 8;
    int dst = sDst[m];
    if (dst >= 0)
      atomicAdd(&h2[(size_t)dst * HID + col], acc[v] * sScale[m]);
  }
}

// ---------------- out = relu(h2) @ lin_w + lin_b (C padded 40 -> 48) -------
__global__ __launch_bounds__(96) void k_final(const float* __restrict__ h2,
                                              const float* __restrict__ lin_w,
                                              const float* __restrict__ lin_b,
                                              float* __restrict__ out) {
  __shared__ float As[16][HID];
  int row0 = blockIdx.x * 16;
  int tid = threadIdx.x;
  for (int idx = tid; idx < 16 * HID; idx += 96) {
    int m = idx >> 7, kk = idx & (HID - 1);
    float v = h2[(size_t)(row0 + m) * HID + kk];
    As[m][kk] = v > 0.f ? v : 0.f;                 // fused relu
  }
  __syncthreads();
  int wave = tid >> 5, lane = tid & 31;
  int half = lane >> 4, ln = lane & 15;
  int col = wave * 16 + ln;                        // 0..47, valid < 40
  v8f acc = {};
  for (int k0 = 0; k0 < HID; k0 += 4) {
    int ka = k0 + half * 2;
    v2f a, b;
    a[0] = As[ln][ka];  a[1] = As[ln][ka + 1];
    b[0] = (col < NCLS) ? lin_w[(size_t)ka * NCLS + col] : 0.f;
    b[1] = (col < NCLS) ? lin_w[(size_t)(ka + 1) * NCLS + col] : 0.f;
    acc = wmma_f32_16x16x4(a, b, acc);             // full-wave EXEC here
  }
  if (col < NCLS) {
    float bias = lin_b[col];
    for (int v = 0; v < 8; ++v) {
      int m = v + half * 8;
      out[(size_t)(row0 + m) * NCLS + col] = acc[v] + bias;
    }
  }
}

// ---------------- launch ----------------

extern "C" void kernel_launch(void* const* d_in, const int* in_sizes, int n_in,
                              void* d_out, int out_size, void* d_ws, size_t ws_size,
                              hipStream_t stream) {
  (void)in_sizes; (void)n_in; (void)out_size; (void)ws_size;
  const int*   edge_index = (const int*)d_in[0];   // [2, E]
  const int*   edge_type  = (const int*)d_in[1];   // [E]
  const float* W1    = (const float*)d_in[2];      // [R, N, H]
  const float* root1 = (const float*)d_in[3];      // [N, H]
  const float* b1    = (const float*)d_in[4];      // [H]
  const float* W2    = (const float*)d_in[5];      // [R, H, H]
  const float* root2 = (const float*)d_in[6];      // [H, H]
  const float* b2    = (const float*)d_in[7];      // [H]
  const float* lin_w = (const float*)d_in[8];      // [H, C]
  const float* lin_b = (const float*)d_in[9];      // [C]
  float* out = (float*)d_out;                      // [N, C]

  char* ws = (char*)d_ws;
  // workspace layout (~56 MB total)
  int*   cnt     = (int*)(ws + 0);                 // N*R      (1.6 MB)
  float* h       = (float*)(ws + 1600000);         // N*H      (25.6 MB)
  float* h2      = (float*)(ws + 27200000);        // N*H      (25.6 MB)
  int*   perm    = (int*)(ws + 52800000);          // E + 16R  (3.2 MB)
  int*   relCnt  = (int*)(ws + 56000512);          // 8
  int*   relFill = relCnt + 8;                     // 8
  int*   relOff  = relCnt + 16;                    // 9

  hipMemsetAsync(cnt, 0, (size_t)N_NODES * N_REL * sizeof(int), stream);
  hipMemsetAsync(h, 0, (size_t)N_NODES * HID * sizeof(float), stream);
  hipMemsetAsync(relCnt, 0, 16 * sizeof(int), stream);      // relCnt + relFill
  hipMemsetAsync(perm, 0xFF, (size_t)(N_EDGE + 16 * N_REL) * sizeof(int), stream);

  k_count<<<(N_EDGE + 255) / 256, 256, 0, stream>>>(edge_index, edge_type, cnt, relCnt);
  k_prefix<<<1, 1, 0, stream>>>(relCnt, relOff);
  k_perm<<<(N_EDGE + 255) / 256, 256, 0, stream>>>(edge_type, relOff, relFill, perm);

  // layer 1: one wave per edge -> E*32 threads
  k_l1_scatter<<<(N_EDGE * 32) / 256, 256, 0, stream>>>(edge_index, edge_type, W1, cnt, h);
  k_relu1<<<(N_NODES * HID + 255) / 256, 256, 0, stream>>>(h, root1, b1);

  // layer 2
  k_root_gemm<<<N_NODES / 16, 256, 0, stream>>>(h, root2, b2, h2);
  int maxTiles = (N_EDGE + 16 * N_REL) / 16;       // 50008 (covers worst padding)
  k_edge_gemm<<<maxTiles, 256, 0, stream>>>(h, W2, perm, relOff, cnt,
                                            edge_index, edge_type, h2);

  // classifier
  k_final<<<N_NODES / 16, 96, 0, stream>>>(h2, lin_w, lin_b, out);
}